// Gator_601295422062
// MI455X (gfx1250) — compile-verified
//
#include <hip/hip_runtime.h>
#include <hip/hip_bf16.h>
#include <stdint.h>

// Logic-gate network (LUT network) inference, bit-sliced over the batch.
// B=8192 batch rows -> 256 columns of 32 rows; each column's full 10240-wire
// packed activation vector (uint32 per wire, bit = batch lane) lives in LDS.
// CDNA5 paths: global_load_async_to_lds_b32 (+ s_wait_asynccnt), ds_swizzle
// wave32 butterfly bit-transpose, global_prefetch_b8.

#define BATCH   8192
#define IN_W    2048
#define NROWS   8
#define NGATES  1024
#define OUT_W   512
#define TOTAL_W (IN_W + NROWS * NGATES)   // 10240 wires -> 40 KB LDS
#define THREADS 1024

// One butterfly stage of the 32x32 bit transpose across lanes.
// IMM = ds_swizzle BITMASK_PERM xor pattern (and_mask=0x1f, xor_mask=J).
// keep = bits this lane keeps; partner bits arrive shifted by J.
#define XSTAGE(w, J, LO, IMM)                                                  \
    do {                                                                       \
        unsigned p_  = (unsigned)__builtin_amdgcn_ds_swizzle((int)(w), (IMM)); \
        unsigned up_ = lane & (unsigned)(J);                                   \
        unsigned ps_ = up_ ? (p_ >> (J)) : (p_ << (J));                        \
        unsigned km_ = up_ ? ((LO) << (J)) : (LO);                             \
        (w) = ((w) & km_) | (ps_ & ~km_); /* -> v_bfi_b32 */                   \
    } while (0)

__global__ __launch_bounds__(THREADS)
void gator_lut_kernel(const unsigned char* __restrict__ x,        // [B, IN] bool (1B)
                      const unsigned char* __restrict__ gates,    // [R, G, 4] bool
                      const int*           __restrict__ choices,  // [R, G, 2] int32
                      const int*           __restrict__ outg,     // [OUT] int32
                      float*               __restrict__ out)      // [B, OUT]
{
    __shared__ unsigned int acts[TOTAL_W]; // packed activations, 40 KB
    __shared__ int og[OUT_W];              // staged output_gates, 2 KB

    const unsigned tid   = threadIdx.x;
    const unsigned lane  = tid & 31u;
    const unsigned wv    = tid >> 5;          // wave id 0..31 (wave32)
    const unsigned col   = blockIdx.x;        // batch-word column 0..255
    const unsigned bbase = col * 32u;

    // ---- Async stage output_gates -> LDS (ASYNCcnt path); overlaps all compute.
    if (tid < OUT_W) {
        unsigned ldsa = (unsigned)(uint64_t)(uintptr_t)&og[tid]; // low 32b of flat LDS addr == LDS offset
        unsigned voff = tid * 4u;
        unsigned long long gp = (unsigned long long)(uintptr_t)outg;
        asm volatile("global_load_async_to_lds_b32 %0, %1, %2"
                     :: "v"(ldsa), "v"(voff), "s"(gp) : "memory");
    }

    // ---- Pack input bits: wave wv owns wires [wv*64, wv*64+64), two 32-wire
    // blocks. Each lane loads 64 contiguous bytes of its own batch row (4x
    // b128), compresses bool bytes to bits with the 0x01020408 multiply trick,
    // then a 5-stage ds_swizzle butterfly transposes the 32x32 bit block so
    // lane l holds wire (base+l) packed over 32 batch rows. Two coalesced
    // ds_store_b32 per wave.
    {
        const unsigned wire0 = wv * 64u;
        const uint4* xrow = (const uint4*)(x + (size_t)(bbase + lane) * IN_W + wire0);
#pragma unroll
        for (int blk = 0; blk < 2; ++blk) {
            uint4 qa = xrow[2 * blk + 0];
            uint4 qb = xrow[2 * blk + 1];
            unsigned d[8] = {qa.x, qa.y, qa.z, qa.w, qb.x, qb.y, qb.z, qb.w};
            unsigned w = 0u;
#pragma unroll
            for (int k = 0; k < 8; ++k) {
                unsigned nib = ((d[k] & 0x01010101u) * 0x01020408u) >> 24; // 4 bools -> nibble
                w |= nib << (4 * k);                 // bit c = x[row lane][wire base+c]
            }
            XSTAGE(w, 16, 0x0000FFFFu, 0x401F);      // SWAPX16
            XSTAGE(w,  8, 0x00FF00FFu, 0x201F);      // SWAPX8
            XSTAGE(w,  4, 0x0F0F0F0Fu, 0x101F);      // SWAPX4
            XSTAGE(w,  2, 0x33333333u, 0x081F);      // SWAPX2
            XSTAGE(w,  1, 0x55555555u, 0x041F);      // SWAPX1
            acts[wire0 + 32u * (unsigned)blk + lane] = w;   // coalesced
        }
    }
    __syncthreads();

    // ---- 8 gate rows; thread g evaluates gate g. Writes target fresh wires
    // (choices[r] indices < IN + r*G), so only one barrier per row is needed.
    for (int r = 0; r < NROWS; ++r) {
        if (r + 1 < NROWS) { // warm next row's params (global_prefetch_b8)
            __builtin_prefetch(choices + (size_t)(r + 1) * NGATES * 2 + tid * 2, 0, 1);
            __builtin_prefetch(gates   + (size_t)(r + 1) * NGATES * 4 + tid * 4, 0, 1);
        }
        const int2 c  = ((const int2*)(choices + (size_t)r * NGATES * 2))[tid];
        unsigned  gb  = ((const unsigned*)(gates + (size_t)r * NGATES * 4))[tid];
        unsigned  a   = acts[c.x];
        unsigned  b   = acts[c.y];
        // truth-table bytes -> full-word masks (shl+ashr sign extend)
        unsigned m0 = (unsigned)((int)(gb << 31) >> 31);
        unsigned m1 = (unsigned)((int)(gb << 23) >> 31);
        unsigned m2 = (unsigned)((int)(gb << 15) >> 31);
        unsigned m3 = (unsigned)((int)(gb <<  7) >> 31);
        // res = mux(b, mux(a, m3, m1), mux(a, m2, m0))  -> 3x v_bfi/bitop3
        unsigned hi  = (a & m3) | (~a & m1);
        unsigned lo  = (a & m2) | (~a & m0);
        unsigned res = (b & hi) | (~b & lo);
        acts[IN_W + r * NGATES + tid] = res;
        __syncthreads();
    }

    // ---- Drain async copy, then everyone can read og from LDS.
    asm volatile("s_wait_asynccnt 0x0" ::: "memory");
    __syncthreads();

    // ---- Gather + unpack to float. Thread t owns 4 consecutive output gates
    // (hoists its 4 LDS gathers once) and 4 batch lanes -> 4 coalesced b128 stores.
    {
        const unsigned ogrp = (tid & 127u) * 4u;   // output-gate group base
        const unsigned blb  = tid >> 7;            // batch lane base 0..7
        const unsigned wa = acts[og[ogrp + 0u]];
        const unsigned wb = acts[og[ogrp + 1u]];
        const unsigned wc = acts[og[ogrp + 2u]];
        const unsigned wd = acts[og[ogrp + 3u]];
#pragma unroll
        for (int k = 0; k < 4; ++k) {
            const unsigned bl = blb + 8u * (unsigned)k;   // 0..31
            float4 f;
            f.x = (float)((wa >> bl) & 1u);
            f.y = (float)((wb >> bl) & 1u);
            f.z = (float)((wc >> bl) & 1u);
            f.w = (float)((wd >> bl) & 1u);
            *(float4*)(out + (size_t)(bbase + bl) * OUT_W + ogrp) = f;
        }
    }
}

extern "C" void kernel_launch(void* const* d_in, const int* in_sizes, int n_in,
                              void* d_out, int out_size, void* d_ws, size_t ws_size,
                              hipStream_t stream) {
    (void)in_sizes; (void)n_in; (void)out_size; (void)d_ws; (void)ws_size;
    const unsigned char* x       = (const unsigned char*)d_in[0];
    const unsigned char* gates   = (const unsigned char*)d_in[1];
    const int*           choices = (const int*)d_in[2];
    const int*           outg    = (const int*)d_in[3];
    float*               out     = (float*)d_out;

    dim3 grid(BATCH / 32);   // 256 columns
    dim3 block(THREADS);     // 32 waves (wave32)
    hipLaunchKernelGGL(gator_lut_kernel, grid, block, 0, stream,
                       x, gates, choices, outg, out);
}